// Attn_30391188586929
// MI455X (gfx1250) — compile-verified
//
#include <hip/hip_runtime.h>
#include <hip/hip_bf16.h>
#include <math.h>

// Problem constants (from reference)
#define H2  1024
#define SEQ 65536
#define NPB 64          // partial-reduction blocks (SEQ / 1024)

typedef __attribute__((ext_vector_type(2))) float v2f;
typedef __attribute__((ext_vector_type(4))) float v4f;
typedef __attribute__((ext_vector_type(8))) float v8f;

// ---------------------------------------------------------------------------
// Kernel 1: u[h] = sum_o W[o][h] * v[o]   (W is [H2, H2] row-major)
// 4 MB read, trivially memory bound; coalesced across h, v[o] is uniform.
// ---------------------------------------------------------------------------
__global__ __launch_bounds__(256) void k_wt_v(const float* __restrict__ W,
                                              const float* __restrict__ v,
                                              float* __restrict__ u) {
    const int h = blockIdx.x * 256 + threadIdx.x;
    float acc = 0.0f;
    #pragma unroll 8
    for (int o = 0; o < H2; ++o) {
        acc = fmaf(W[(size_t)o * H2 + h], v[o], acc);
    }
    u[h] = acc;
}

// ---------------------------------------------------------------------------
// Kernel 2: e[s] = dot(outputs[s, :], u)  via V_WMMA_F32_16X16X4_F32.
//
// Each wave computes 16 rows. Per K-step of 4:
//   A (16x4 fp32, 2 VGPRs): lanes 0-15 hold row m=lane, K = k0+0..1 ;
//                           lanes 16-31 hold row m=lane-16, K = k0+2..3.
//   B (4x16 fp32, 2 VGPRs): u[k] broadcast across all 16 N columns — the lane
//     index pattern mirrors A, so the same (k0 + 2*(lane>>4)) offset works.
// All D columns then hold the identical dot result: lane 0's c[0..7] are rows
// 0..7, lane 16's c[0..7] are rows 8..15.
//
// outputs (256 MB, > 192 MB L2, touched once) is streamed non-temporally.
// 8 waves / block -> 128 rows per block -> 512 blocks.
// ---------------------------------------------------------------------------
__global__ __launch_bounds__(256) void k_gemv_wmma(const float* __restrict__ X,
                                                   const float* __restrict__ u,
                                                   float* __restrict__ e) {
    __shared__ float su[H2];
    // one b128 per thread stages all of u (1024 floats / 256 threads)
    ((v4f*)su)[threadIdx.x] = ((const v4f*)u)[threadIdx.x];
    __syncthreads();

    const int wave    = threadIdx.x >> 5;           // 0..7
    const int lane    = threadIdx.x & 31;
    const int rowBase = blockIdx.x * 128 + wave * 16;
    const int m       = lane & 15;                  // row within tile
    const int khalf   = (lane >> 4) << 1;           // 0 or 2 (K sub-offset)

    const float* rowPtr = X + (size_t)(rowBase + m) * H2 + khalf;
    const float* uPtr   = su + khalf;

    v8f c = {};                                     // fp32 accumulator tile
    #pragma unroll 8
    for (int k0 = 0; k0 < H2; k0 += 4) {
        v2f a = __builtin_nontemporal_load((const v2f*)(rowPtr + k0));
        v2f b = *(const v2f*)(uPtr + k0);
        // (neg_a, A, neg_b, B, c_mod, C, reuse_a, reuse_b)
        c = __builtin_amdgcn_wmma_f32_16x16x4_f32(false, a, false, b,
                                                  (short)0, c, false, false);
    }

    // Every column of D is identical: extract the 16 row results.
    if (lane == 0) {
        #pragma unroll
        for (int r = 0; r < 8; ++r) e[rowBase + r] = c[r];
    } else if (lane == 16) {
        #pragma unroll
        for (int r = 0; r < 8; ++r) e[rowBase + 8 + r] = c[r];
    }
}

// ---------------------------------------------------------------------------
// Kernel 3: parallel online-softmax partials. Block j owns a contiguous 1024-
// element slice: computes m_j = max(slice), s_j = sum(exp(x - m_j)) with
// fixed-tree (deterministic) reductions. 64 blocks -> full-device bandwidth.
// ---------------------------------------------------------------------------
__global__ __launch_bounds__(256) void k_partials(const float* __restrict__ e,
                                                  float* __restrict__ pmax,
                                                  float* __restrict__ psum) {
    __shared__ float red[256];
    const int t = threadIdx.x;
    const v4f x = *(const v4f*)(e + blockIdx.x * 1024 + t * 4);

    float m = fmaxf(fmaxf(x[0], x[1]), fmaxf(x[2], x[3]));
    red[t] = m;
    __syncthreads();
    for (int s = 128; s > 0; s >>= 1) {
        if (t < s) red[t] = fmaxf(red[t], red[t + s]);
        __syncthreads();
    }
    const float bm = red[0];
    __syncthreads();

    float sum = __expf(x[0] - bm) + __expf(x[1] - bm) +
                __expf(x[2] - bm) + __expf(x[3] - bm);
    red[t] = sum;
    __syncthreads();
    for (int s = 128; s > 0; s >>= 1) {
        if (t < s) red[t] += red[t + s];
        __syncthreads();
    }
    if (t == 0) {
        pmax[blockIdx.x] = bm;
        psum[blockIdx.x] = red[0];
    }
}

// ---------------------------------------------------------------------------
// Kernel 4: combine the 64 (m_j, s_j) partials:
//   gmax = max_j m_j ;  gsum = sum_j s_j * exp(m_j - gmax)
// Single tiny block, fixed reduction tree (deterministic).
// ---------------------------------------------------------------------------
__global__ __launch_bounds__(64) void k_combine(const float* __restrict__ pmax,
                                                const float* __restrict__ psum,
                                                float* __restrict__ stats) {
    __shared__ float rm[NPB];
    __shared__ float rs[NPB];
    const int t = threadIdx.x;
    const float mj = pmax[t];
    rm[t] = mj;
    __syncthreads();
    for (int s = NPB / 2; s > 0; s >>= 1) {
        if (t < s) rm[t] = fmaxf(rm[t], rm[t + s]);
        __syncthreads();
    }
    const float gmax = rm[0];
    __syncthreads();

    rs[t] = psum[t] * __expf(mj - gmax);
    __syncthreads();
    for (int s = NPB / 2; s > 0; s >>= 1) {
        if (t < s) rs[t] += rs[t + s];
        __syncthreads();
    }
    if (t == 0) {
        stats[0] = gmax;
        stats[1] = rs[0];
    }
}

// ---------------------------------------------------------------------------
// Kernel 5: out[s] = exp(e[s] - gmax) * (1/gsum), vectorized b128 in/out.
// ---------------------------------------------------------------------------
__global__ __launch_bounds__(256) void k_normalize(const float* __restrict__ e,
                                                   const float* __restrict__ stats,
                                                   float* __restrict__ out) {
    const int i = (blockIdx.x * 256 + threadIdx.x) * 4;
    const float gmax = stats[0];
    const float rsum = 1.0f / stats[1];
    const v4f x = *(const v4f*)(e + i);
    v4f r;
    r[0] = __expf(x[0] - gmax) * rsum;
    r[1] = __expf(x[1] - gmax) * rsum;
    r[2] = __expf(x[2] - gmax) * rsum;
    r[3] = __expf(x[3] - gmax) * rsum;
    *(v4f*)(out + i) = r;
}

// ---------------------------------------------------------------------------
// launch: u = W^T v  ->  e = outputs @ u  ->  partial softmax -> combine ->
// normalize. Bias b dropped: softmax(x + const) == softmax(x).
// Workspace: [u: 1024 f][e: 65536 f][pmax: 64 f][psum: 64 f][stats: 2 f]
// ---------------------------------------------------------------------------
extern "C" void kernel_launch(void* const* d_in, const int* in_sizes, int n_in,
                              void* d_out, int out_size, void* d_ws, size_t ws_size,
                              hipStream_t stream) {
    const float* outputs    = (const float*)d_in[0];   // [SEQ, H2]
    const float* W          = (const float*)d_in[1];   // [H2, H2]
    const float* b          = (const float*)d_in[2];   // [H2] (unused)
    const float* weight_vec = (const float*)d_in[3];   // [H2]
    (void)b; (void)in_sizes; (void)n_in; (void)out_size; (void)ws_size;

    float* u     = (float*)d_ws;        // 1024 floats
    float* e     = u + H2;              // 65536 floats
    float* pmax  = e + SEQ;             // 64 floats
    float* psum  = pmax + NPB;          // 64 floats
    float* stats = psum + NPB;          // 2 floats
    float* out   = (float*)d_out;       // 65536 floats

    k_wt_v      <<<H2 / 256,    256, 0, stream>>>(W, weight_vec, u);
    k_gemv_wmma <<<SEQ / 128,   256, 0, stream>>>(outputs, u, e);
    k_partials  <<<NPB,         256, 0, stream>>>(e, pmax, psum);
    k_combine   <<<1,           NPB, 0, stream>>>(pmax, psum, stats);
    k_normalize <<<SEQ / 1024,  256, 0, stream>>>(e, stats, out);
}